// WRGAT_2370821947939
// MI455X (gfx1250) — compile-verified
//
#include <hip/hip_runtime.h>
#include <hip/hip_bf16.h>

// ---------------- problem constants ----------------
#define NN   100000      // nodes
#define NE   1600000     // edges
#define FDIM 128         // input features
#define DDIM 64          // hidden dims
#define CDIM 16          // classes
#define RREL 8           // relations
#define NEG_SLOPE 0.2f

typedef float v2f __attribute__((ext_vector_type(2)));
typedef float v8f __attribute__((ext_vector_type(8)));

// ---------------------------------------------------------------------------
// WMMA f32 16x16x4 GEMM:  D[m,n] = sum_k A[m,k]*B[k,n] (+Cin) (+bias) (relu?)
// A: [M,K] row-major, B: [K,Nd] row-major (per-relation slab via blockIdx.y),
// D: [M,Nd] row-major (per-relation slab via blockIdx.y).
// One wave computes one 16x16 tile; 8 waves (256 threads) per block.
// Bounds check is wave-uniform so EXEC is all-ones inside WMMA waves.
// ---------------------------------------------------------------------------
__global__ void wmma_gemm_f32(const float* __restrict__ A,
                              const float* __restrict__ B,
                              const float* __restrict__ Cin,
                              const float* __restrict__ bias,
                              float* __restrict__ Dmat,
                              int M, int K, int Nd, int relu) {
  B    += (size_t)blockIdx.y * K * Nd;
  Dmat += (size_t)blockIdx.y * M * Nd;

  const int lane   = threadIdx.x & 31;
  const int wave   = threadIdx.x >> 5;
  const int ntiles = Nd >> 4;
  const int mtiles = M >> 4;
  const int gw     = blockIdx.x * (blockDim.x >> 5) + wave;
  if (gw >= mtiles * ntiles) return;          // wave-uniform

  const int mt = gw / ntiles, nt = gw % ntiles;
  const int m0 = mt << 4, n0 = nt << 4;
  const int half  = lane >> 4;                // selects K pair / M upper-half
  const int idx15 = lane & 15;                // A row / B,C,D column

  v8f acc;
  if (Cin) {
#pragma unroll
    for (int i = 0; i < 8; ++i)
      acc[i] = Cin[(size_t)(m0 + i + 8 * half) * Nd + n0 + idx15];
  } else {
#pragma unroll
    for (int i = 0; i < 8; ++i) acc[i] = 0.0f;
  }

  const float* Arow = A + (size_t)(m0 + idx15) * K;
  for (int k = 0; k < K; k += 4) {
    v2f a, b;
    // A 16x4 layout: lanes 0-15 hold K={0,1}, lanes 16-31 hold K={2,3}
    a.x = Arow[k + 2 * half + 0];
    a.y = Arow[k + 2 * half + 1];
    // B 4x16 layout: VGPR v, lanes 0-15 -> K=v, lanes 16-31 -> K=v+2
    const float* Bp = B + (size_t)(k + 2 * half) * Nd + n0 + idx15;
    b.x = Bp[0];
    b.y = Bp[Nd];
    acc = __builtin_amdgcn_wmma_f32_16x16x4_f32(
        false, a, false, b, (short)0, acc, false, false);
  }

  const float bv = bias ? bias[n0 + idx15] : 0.0f;
#pragma unroll
  for (int i = 0; i < 8; ++i) {
    float v = acc[i] + bv;
    if (relu) v = v > 0.0f ? v : 0.0f;
    Dmat[(size_t)(m0 + i + 8 * half) * Nd + n0 + idx15] = v;
  }
}

// ---------------------------------------------------------------------------
// Per-(relation,node) attention scalars:
//   sdst[r*NN+n] = dot(h[r,n,:], att[r, 0:dim])     (a_dst = first half)
//   ssrc[r*NN+n] = dot(h[r,n,:], att[r, dim:2dim])  (a_src = second half)
// One wave per (r,n); lanes stride features, reduce via shfl_xor.
// ---------------------------------------------------------------------------
__global__ void node_att_scalars(const float* __restrict__ h,
                                 const float* __restrict__ att,
                                 float* __restrict__ sdst,
                                 float* __restrict__ ssrc,
                                 int dim) {
  const int gwave = (int)((blockIdx.x * blockDim.x + threadIdx.x) >> 5);
  const int lane  = threadIdx.x & 31;
  if (gwave >= RREL * NN) return;
  const int r = gwave / NN;
  const float* hv = h + (size_t)gwave * dim;
  const float* ad = att + (size_t)r * 2 * dim;
  const float* as = ad + dim;
  float pd = 0.0f, ps = 0.0f;
  for (int f = lane; f < dim; f += 32) {
    const float xv = hv[f];
    pd += xv * ad[f];
    ps += xv * as[f];
  }
#pragma unroll
  for (int off = 16; off; off >>= 1) {
    pd += __shfl_xor(pd, off, 32);
    ps += __shfl_xor(ps, off, 32);
  }
  if (lane == 0) { sdst[gwave] = pd; ssrc[gwave] = ps; }
}

// order-preserving float->uint key (handles negatives) for atomic max
__device__ __forceinline__ unsigned f32key(float f) {
  unsigned u = __float_as_uint(f);
  return (u & 0x80000000u) ? ~u : (u | 0x80000000u);
}
__device__ __forceinline__ float keyf32(unsigned k) {
  unsigned u = (k & 0x80000000u) ? (k & 0x7FFFFFFFu) : ~k;
  return __uint_as_float(u);
}

// e = leaky_relu(sdst[et,dst] + ssrc[et,src]); segment max via atomicMax(key)
__global__ void edge_score_max(const int* __restrict__ srcI,
                               const int* __restrict__ dstI,
                               const int* __restrict__ et,
                               const float* __restrict__ sdst,
                               const float* __restrict__ ssrc,
                               float* __restrict__ ebuf,
                               unsigned* __restrict__ mkey) {
  const int i = blockIdx.x * blockDim.x + threadIdx.x;
  if (i >= NE) return;
  const int r = et[i], d = dstI[i], s = srcI[i];
  float e = sdst[(size_t)r * NN + d] + ssrc[(size_t)r * NN + s];
  e = e > 0.0f ? e : NEG_SLOPE * e;
  ebuf[i] = e;
  atomicMax(mkey + d, f32key(e));
}

// ex = exp(e - m[dst]); segment sum via atomicAdd; ex stored over ebuf
__global__ void edge_exp_sum(const int* __restrict__ dstI,
                             float* __restrict__ ebuf,
                             const unsigned* __restrict__ mkey,
                             float* __restrict__ ssum) {
  const int i = blockIdx.x * blockDim.x + threadIdx.x;
  if (i >= NE) return;
  const int d = dstI[i];
  const float ex = expf(ebuf[i] - keyf32(mkey[d]));
  ebuf[i] = ex;
  atomicAdd(ssum + d, ex);
}

// agg[dst,:] += (ew * ex / (sum[dst]+eps)) * h[et,src,:]  — one wave per edge
__global__ void edge_scatter(const int* __restrict__ srcI,
                             const int* __restrict__ dstI,
                             const int* __restrict__ et,
                             const float* __restrict__ ew,
                             const float* __restrict__ exbuf,
                             const float* __restrict__ ssum,
                             const float* __restrict__ h,
                             float* __restrict__ agg,
                             int dim) {
  const int gwave = (int)((blockIdx.x * blockDim.x + threadIdx.x) >> 5);
  const int lane  = threadIdx.x & 31;
  if (gwave >= NE) return;
  const int d = dstI[gwave], s = srcI[gwave], r = et[gwave];
  const float w = ew[gwave] * exbuf[gwave] / (ssum[d] + 1e-16f);
  const float* hv = h + ((size_t)r * NN + s) * dim;
  float* ag = agg + (size_t)d * dim;
  for (int f = lane; f < dim; f += 32)
    atomicAdd(ag + f, w * hv[f]);
}

// per-node log-softmax over 16 classes; logits already sit in d_out 2nd half
__global__ void log_softmax16(const float* __restrict__ logits,
                              float* __restrict__ out) {
  const int n = blockIdx.x * blockDim.x + threadIdx.x;
  if (n >= NN) return;
  const float* l = logits + (size_t)n * CDIM;
  float m = l[0];
#pragma unroll
  for (int i = 1; i < CDIM; ++i) m = fmaxf(m, l[i]);
  float sum = 0.0f;
#pragma unroll
  for (int i = 0; i < CDIM; ++i) sum += expf(l[i] - m);
  const float lse = m + logf(sum);
  float* o = out + (size_t)n * CDIM;
#pragma unroll
  for (int i = 0; i < CDIM; ++i) o[i] = l[i] - lse;
}

// ---------------------------------------------------------------------------
extern "C" void kernel_launch(void* const* d_in, const int* in_sizes, int n_in,
                              void* d_out, int out_size, void* d_ws, size_t ws_size,
                              hipStream_t stream) {
  (void)in_sizes; (void)n_in; (void)out_size; (void)ws_size;

  const float* x     = (const float*)d_in[0];
  const int*   ei    = (const int*)  d_in[1];   // [2, NE]
  const float* ew    = (const float*)d_in[2];
  const int*   ec    = (const int*)  d_in[3];
  const float* W1    = (const float*)d_in[4];
  const float* att1  = (const float*)d_in[5];
  const float* root1 = (const float*)d_in[6];
  const float* b1    = (const float*)d_in[7];
  const float* W2    = (const float*)d_in[8];
  const float* att2  = (const float*)d_in[9];
  const float* root2 = (const float*)d_in[10];
  const float* b2    = (const float*)d_in[11];
  const int* srcI = ei;
  const int* dstI = ei + NE;

  float* out = (float*)d_out;
  float* logits = out + (size_t)NN * CDIM;      // second half of output tuple

  // ---- workspace carve-up (fp32 words) ----
  float* ws = (float*)d_ws;
  float*    h1   = ws;                                   // R*NN*D
  float*    s1d  = h1  + (size_t)RREL * NN * DDIM;       // R*NN
  float*    s1s  = s1d + (size_t)RREL * NN;              // R*NN
  float*    ebuf = s1s + (size_t)RREL * NN;              // NE (scores -> exp)
  unsigned* mkey = (unsigned*)(ebuf + NE);               // NN
  float*    ssum = (float*)(mkey + NN);                  // NN
  float*    agg1 = ssum + NN;                            // NN*D
  float*    hid  = agg1 + (size_t)NN * DDIM;             // NN*D
  float*    h2   = hid  + (size_t)NN * DDIM;             // R*NN*C
  float*    s2d  = h2   + (size_t)RREL * NN * CDIM;      // R*NN
  float*    s2s  = s2d  + (size_t)RREL * NN;             // R*NN
  float*    agg2 = s2s  + (size_t)RREL * NN;             // NN*C

  const int THREADS = 256;
  const int gemmBlk1  = (NN / 16) * (DDIM / 16) / 8;           // 3125
  const int gemmBlk2  = ((NN / 16) * (CDIM / 16) + 7) / 8;     // 782
  const int scalBlk   = RREL * NN / 8;                         // 100000 (wave/(r,n))
  const int edgeBlk   = (NE + THREADS - 1) / THREADS;          // 6250
  const int scatBlk   = NE / 8;                                // 200000 (wave/edge)
  const int lsmBlk    = (NN + THREADS - 1) / THREADS;          // 391

  // ---- init segment-reduction buffers ----
  hipMemsetAsync(mkey, 0, (size_t)NN * 4, stream);             // key 0 == -inf
  hipMemsetAsync(ssum, 0, (size_t)NN * 4, stream);
  hipMemsetAsync(agg1, 0, (size_t)NN * DDIM * 4, stream);
  hipMemsetAsync(agg2, 0, (size_t)NN * CDIM * 4, stream);

  // ================= layer 1 (in=128, out=64) =================
  wmma_gemm_f32<<<dim3(gemmBlk1, RREL), THREADS, 0, stream>>>(
      x, W1, nullptr, nullptr, h1, NN, FDIM, DDIM, 0);
  node_att_scalars<<<scalBlk, THREADS, 0, stream>>>(h1, att1, s1d, s1s, DDIM);
  edge_score_max<<<edgeBlk, THREADS, 0, stream>>>(srcI, dstI, ec, s1d, s1s, ebuf, mkey);
  edge_exp_sum<<<edgeBlk, THREADS, 0, stream>>>(dstI, ebuf, mkey, ssum);
  edge_scatter<<<scatBlk, THREADS, 0, stream>>>(srcI, dstI, ec, ew, ebuf, ssum,
                                                h1, agg1, DDIM);
  // hid = relu(agg1 + x @ root1 + b1)
  wmma_gemm_f32<<<dim3(gemmBlk1, 1), THREADS, 0, stream>>>(
      x, root1, agg1, b1, hid, NN, FDIM, DDIM, 1);

  // ================= layer 2 (in=64, out=16) =================
  hipMemsetAsync(mkey, 0, (size_t)NN * 4, stream);
  hipMemsetAsync(ssum, 0, (size_t)NN * 4, stream);

  wmma_gemm_f32<<<dim3(gemmBlk2, RREL), THREADS, 0, stream>>>(
      hid, W2, nullptr, nullptr, h2, NN, DDIM, CDIM, 0);
  node_att_scalars<<<scalBlk, THREADS, 0, stream>>>(h2, att2, s2d, s2s, CDIM);
  edge_score_max<<<edgeBlk, THREADS, 0, stream>>>(srcI, dstI, ec, s2d, s2s, ebuf, mkey);
  edge_exp_sum<<<edgeBlk, THREADS, 0, stream>>>(dstI, ebuf, mkey, ssum);
  edge_scatter<<<scatBlk, THREADS, 0, stream>>>(srcI, dstI, ec, ew, ebuf, ssum,
                                                h2, agg2, CDIM);
  // logits = agg2 + hid @ root2 + b2   (written straight into d_out 2nd half)
  wmma_gemm_f32<<<dim3(gemmBlk2, 1), THREADS, 0, stream>>>(
      hid, root2, agg2, b2, logits, NN, DDIM, CDIM, 0);

  // d_out first half = log_softmax(logits)
  log_softmax16<<<lsmBlk, THREADS, 0, stream>>>(logits, out);
}